// sub_graph_parallel_38250978738662
// MI455X (gfx1250) — compile-verified
//
#include <hip/hip_runtime.h>

typedef float v2f __attribute__((ext_vector_type(2)));
typedef float v8f __attribute__((ext_vector_type(8)));

#define Bb   16
#define Nn   2000
#define NSs  20
#define FIN  64
#define FOUT 64
#define NT   4
#define ROWS (Bb*Nn)          // 32000

// workspace layout (float offsets)
#define WS_RSUM   0           // Nn*NSs          = 40000
#define WS_A      40000       // NSs*FOUT        = 1280
#define WS_C      41280       // NSs*FOUT        = 1280
#define WS_AW     42560       // NT*NSs*FOUT     = 5120
#define WS_CW     47680       // 5120
#define WS_BW     52800       // 5120
#define WS_MFULL  57920       // 160*64 = 10240 (pair-interleaved, 16B-aligned base)
#define WS_PBP    68160       // NT*NSs*FOUT     = 5120
#define WS_PBS    73280       // NT*FOUT         = 256

// pair-interleaved index for B_full element [kg][c], kg in [0,160), c in [0,64):
// group g = kg>>2 (4 K-rows per WMMA step), half hh = (kg>>1)&1, odd o = kg&1.
// A lane (half hh, column c) reads the pair at v2f index (g*2+hh)*64 + c.
__device__ __forceinline__ int mf_idx(int kg, int c) {
    int g = kg >> 2, hh = (kg >> 1) & 1, o = kg & 1;
    return (((g * 2 + hh) * 64) + c) * 2 + o;
}

// ---------------- precompute kernels (tiny, scalar f32) ----------------

// r[n,s] = sum_j adj_sub[n,s,j]
__global__ void k_rsum(const float* __restrict__ adj, float* __restrict__ rsum) {
    int idx = blockIdx.x * 256 + threadIdx.x;
    if (idx >= Nn * NSs) return;
    const float* p = adj + idx * NSs;
    float s = 0.f;
    #pragma unroll
    for (int j = 0; j < NSs; ++j) s += p[j];
    rsum[idx] = s;
}

// A[s,o] = sum_f lin_in_w[f]*weight[s,f,o]; C[s,o] = sum_f lin_in_b[f]*weight[s,f,o]
__global__ void k_AC(const float* __restrict__ weight, const float* __restrict__ w_in,
                     const float* __restrict__ b_in, float* __restrict__ A, float* __restrict__ C) {
    int s = blockIdx.x, o = threadIdx.x;
    float aa = 0.f, cc = 0.f;
    for (int f = 0; f < FIN; ++f) {
        float w = weight[(s * FIN + f) * FOUT + o];
        aa += w_in[f] * w;
        cc += b_in[f] * w;
    }
    A[s * FOUT + o] = aa;
    C[s * FOUT + o] = cc;
}

// AW/CW/BW[t,s,m] = sum_o {A,C,bias}[s,o] * W_subgraphs[t,o,m]
__global__ void k_TW(const float* __restrict__ Wsub, const float* __restrict__ A,
                     const float* __restrict__ C, const float* __restrict__ bias,
                     float* __restrict__ AW, float* __restrict__ CW, float* __restrict__ BW) {
    int t = blockIdx.x / NSs, s = blockIdx.x % NSs, m = threadIdx.x;
    float aw = 0.f, cw = 0.f, bw = 0.f;
    for (int o = 0; o < FOUT; ++o) {
        float w = Wsub[(t * FOUT + o) * FOUT + m];
        aw += A[s * FOUT + o] * w;
        cw += C[s * FOUT + o] * w;
        bw += bias[s * FOUT + o] * w;
    }
    int idx = (t * NSs + s) * FOUT + m;
    AW[idx] = aw; CW[idx] = cw; BW[idx] = bw;
}

// B_full row (t*40+s)    [mp] = sum_m AW[t,s,m]*lin_w[mp, s*64+m]
// B_full row (t*40+20+s) [mp] = sum_m CW[t,s,m]*lin_w[mp, s*64+m]   (pair-interleaved)
// PbP[t,s,mp]                 = sum_m BW[t,s,m]*lin_w[mp, s*64+m]
__global__ void k_P(const float* __restrict__ AW, const float* __restrict__ CW,
                    const float* __restrict__ BW, const float* __restrict__ lin_w,
                    float* __restrict__ Mfull, float* __restrict__ PbP) {
    int t = blockIdx.x / NSs, s = blockIdx.x % NSs, mp = threadIdx.x;
    float p = 0.f, pc = 0.f, pb = 0.f;
    for (int m = 0; m < FOUT; ++m) {
        float l = lin_w[mp * (NSs * FOUT) + s * FOUT + m];
        int idx = (t * NSs + s) * FOUT + m;
        p  += AW[idx] * l;
        pc += CW[idx] * l;
        pb += BW[idx] * l;
    }
    Mfull[mf_idx(t * 40 + s,      mp)] = p;
    Mfull[mf_idx(t * 40 + 20 + s, mp)] = pc;
    PbP[(t * NSs + s) * 64 + mp] = pb;
}

// PbS[t,mp] = lin_b[mp] + sum_s PbP[t,s,mp]
__global__ void k_PbS(const float* __restrict__ PbP, const float* __restrict__ lin_b,
                      float* __restrict__ PbS) {
    int t = blockIdx.x, mp = threadIdx.x;
    float s = lin_b[mp];
    #pragma unroll
    for (int k = 0; k < NSs; ++k) s += PbP[(t * NSs + k) * 64 + mp];
    PbS[t * 64 + mp] = s;
}

// ---------------- main WMMA kernel ----------------
// Per wave: 16 rows. out_tile(16x64) = A_ext(16x160) @ B_full(160x64) + PbS[t_row]
// A_ext[row, t*40+k] = (t==t_row) ? v[row][k] : 0,  v = [r*x (20), r (20)]
__launch_bounds__(256)
__global__ void k_main_wmma(const float* __restrict__ x, const int* __restrict__ acc_t,
                            const float* __restrict__ rsum, const float* __restrict__ Mfull,
                            const float* __restrict__ PbS, float* __restrict__ out) {
    __shared__ v2f   lds_M[80 * 64];       // B_full pairs, 40 KB
    __shared__ float lds_v[8][16][40];     // per-wave row vectors
    __shared__ int   lds_t[8][16];         // per-row type
    __shared__ float lds_pb[NT * 64];      // bias table

    const int tid = threadIdx.x;
    // contiguous copy: 10240 floats = 2560 float4, 10 per thread (b128 in/out)
    {
        const float4* src = (const float4*)Mfull;
        float4* dst = (float4*)lds_M;
        #pragma unroll
        for (int i = 0; i < 10; ++i) dst[tid + i * 256] = src[tid + i * 256];
    }
    if (tid < NT * 64) lds_pb[tid] = PbS[tid];

    const int wid  = tid >> 5;
    const int lane = tid & 31;
    const int hi   = lane >> 4;     // half-wave select
    const int ri   = lane & 15;     // A-row within tile / column within col-tile
    const int row_base = blockIdx.x * 128 + wid * 16;
    const int row = row_base + ri;
    const int n   = row % Nn;

    if (hi == 0) {
        for (int s = 0; s < NSs; ++s)
            lds_v[wid][ri][s] = rsum[n * NSs + s] * x[row * NSs + s];
        lds_t[wid][ri] = acc_t[row];
    } else {
        for (int s = 0; s < NSs; ++s)
            lds_v[wid][ri][NSs + s] = rsum[n * NSs + s];
    }
    __syncthreads();

    // Preload this lane's A-fragment pairs: step q needs v[q*4+2*hi .. +1] (8B aligned)
    v2f av[10];
    {
        const v2f* vp = (const v2f*)&lds_v[wid][ri][0];
        #pragma unroll
        for (int q = 0; q < 10; ++q) av[q] = vp[q * 2 + hi];
    }
    const int trow = lds_t[wid][ri];   // type of the A-row this lane supplies
    const int col  = ri;

    v8f acc[4] = {v8f{}, v8f{}, v8f{}, v8f{}};

    for (int tb = 0; tb < NT; ++tb) {           // type block: K rows [tb*40, tb*40+40)
        const bool sel = (tb == trow);           // register select -> v_cndmask
        const int pbase = (tb * 10 * 2 + hi) * 64 + col;  // v2f index of (g=tb*10, this half)
        #pragma unroll
        for (int q = 0; q < 10; ++q) {
            v2f a;
            a.x = sel ? av[q].x : 0.f;
            a.y = sel ? av[q].y : 0.f;
            #pragma unroll
            for (int ct = 0; ct < 4; ++ct) {
                // single ds_load_b64: {B[k][c], B[k+1][c]} for this lane half
                v2f b = lds_M[pbase + q * 128 + ct * 16];
                acc[ct] = __builtin_amdgcn_wmma_f32_16x16x4_f32(
                    false, a, false, b, (short)0, acc[ct], false, false);
            }
        }
    }

    // epilogue: C/D layout — VGPR v, lane l: M = v + 8*hi, N = l%16
    int tM[8];
    #pragma unroll
    for (int v = 0; v < 8; ++v) tM[v] = lds_t[wid][v + 8 * hi];

    #pragma unroll
    for (int ct = 0; ct < 4; ++ct) {
        #pragma unroll
        for (int v = 0; v < 8; ++v) {
            int M = v + 8 * hi;
            out[(row_base + M) * 64 + ct * 16 + col] =
                acc[ct][v] + lds_pb[tM[v] * 64 + ct * 16 + col];
        }
    }
}

// ---------------- launcher ----------------
extern "C" void kernel_launch(void* const* d_in, const int* in_sizes, int n_in,
                              void* d_out, int out_size, void* d_ws, size_t ws_size,
                              hipStream_t stream) {
    const float* x      = (const float*)d_in[0];
    const int*   acct   = (const int*)  d_in[1];
    const float* Wsub   = (const float*)d_in[2];
    const float* adj    = (const float*)d_in[3];
    const float* weight = (const float*)d_in[4];
    const float* bias   = (const float*)d_in[5];
    const float* w_in   = (const float*)d_in[6];
    const float* b_in   = (const float*)d_in[7];
    const float* lin_w  = (const float*)d_in[8];
    const float* lin_b  = (const float*)d_in[9];
    float* out = (float*)d_out;
    float* ws  = (float*)d_ws;

    float* rsum  = ws + WS_RSUM;
    float* A_    = ws + WS_A;
    float* C_    = ws + WS_C;
    float* AW_   = ws + WS_AW;
    float* CW_   = ws + WS_CW;
    float* BW_   = ws + WS_BW;
    float* Mfull = ws + WS_MFULL;
    float* PbP   = ws + WS_PBP;
    float* PbS   = ws + WS_PBS;

    k_rsum<<<(Nn * NSs + 255) / 256, 256, 0, stream>>>(adj, rsum);
    k_AC  <<<NSs,      FOUT, 0, stream>>>(weight, w_in, b_in, A_, C_);
    k_TW  <<<NT * NSs, FOUT, 0, stream>>>(Wsub, A_, C_, bias, AW_, CW_, BW_);
    k_P   <<<NT * NSs, FOUT, 0, stream>>>(AW_, CW_, BW_, lin_w, Mfull, PbP);
    k_PbS <<<NT,       FOUT, 0, stream>>>(PbP, lin_b, PbS);
    k_main_wmma<<<ROWS / 128, 256, 0, stream>>>(x, acct, rsum, Mfull, PbS, out);
}